// RandomShiftsAug_63350767616513
// MI455X (gfx1250) — compile-verified
//
#include <hip/hip_runtime.h>
#include <stdint.h>

#define PAD   4
#define N_IMG 256
#define C_CH  3
#define HH    224
#define WW    224

// 256-thread block = 8 waves; each wave handles 4 consecutive rows of one
// (n,c) image plane. Data moves via the CDNA5 async LDS-DMA path with
// b128 granularity; only the <=4-pixel replicated edges fall back to
// per-element clamped b32 async loads.
__global__ __launch_bounds__(256) void RandomShiftsAug_kernel(
    const float* __restrict__ x,
    const int* __restrict__ shift,
    float* __restrict__ out) {

    __shared__ float tile[32 * WW];   // 8 waves * 4 rows * 224 floats = 28672 B

    const int lane = threadIdx.x & 31;
    const int wave = threadIdx.x >> 5;

    const int n = blockIdx.z;                 // image   (uniform -> scalar regs)
    const int c = blockIdx.y;                 // channel (uniform)
    const int y_base = blockIdx.x * 32 + wave * 4;

    // shift[n,0] = sx, shift[n,1] = sy  (block-uniform -> s_load)
    const int sx = shift[2 * n + 0];
    const int sy = shift[2 * n + 1];
    const int dx = sx - PAD;                  // in [-4, 4]

    const int local_row = lane >> 3;          // 0..3
    const int sub       = lane & 7;           // 0..7

    const int y = y_base + local_row;
    int src_y = y + sy - PAD;
    src_y = src_y < 0 ? 0 : (src_y > HH - 1 ? HH - 1 : src_y);

    const size_t plane = ((size_t)n * C_CH + c) * HH;
    const float* src_row = x   + (plane + (size_t)src_y) * WW;
    float*       dst_row = out + (plane + (size_t)y) * WW;

    // LDS byte offset of this lane's row slice (low 32 bits of generic ptr).
    const unsigned lds_row =
        (unsigned)(uintptr_t)(&tile[wave * 4 * WW]) + (unsigned)(local_row * WW * 4);

    // Phase 1: async DMA gather into LDS, b128 interior / clamped b32 edges.
#pragma unroll
    for (int k = 0; k < 7; ++k) {
        const int jj = sub + k * 8;                   // float4 chunk 0..55
        const unsigned laddr = lds_row + (unsigned)(jj * 16);
        const bool lo_edge = (jj == 0) && (dx < 0);   // only possible at k==0
        const bool hi_edge = (jj == 55) && (dx > 0);  // only possible at k==6
        if (lo_edge || hi_edge) {
#pragma unroll
            for (int t = 0; t < 4; ++t) {
                int sxc = jj * 4 + t + dx;
                sxc = sxc < 0 ? 0 : (sxc > WW - 1 ? WW - 1 : sxc);
                const uint64_t g = (uint64_t)(uintptr_t)(src_row + sxc);
                asm volatile("global_load_async_to_lds_b32 %0, %1, off"
                             :: "v"(laddr + (unsigned)(t * 4)), "v"(g) : "memory");
            }
        } else {
            const uint64_t g = (uint64_t)(uintptr_t)(src_row + jj * 4 + dx);
            asm volatile("global_load_async_to_lds_b128 %0, %1, off"
                         :: "v"(laddr), "v"(g) : "memory");
        }
    }

    // Per-wave: all async loads landed in LDS (each wave only touches its slice).
    asm volatile("s_wait_asynccnt 0" ::: "memory");

    // Phase 2: async DMA the contiguous rows LDS -> global, all b128.
#pragma unroll
    for (int k = 0; k < 7; ++k) {
        const int jj = sub + k * 8;
        const unsigned laddr = lds_row + (unsigned)(jj * 16);
        const uint64_t g = (uint64_t)(uintptr_t)(dst_row + jj * 4);
        asm volatile("global_store_async_from_lds_b128 %0, %1, off"
                     :: "v"(g), "v"(laddr) : "memory");
    }

    asm volatile("s_wait_asynccnt 0" ::: "memory");
}

extern "C" void kernel_launch(void* const* d_in, const int* in_sizes, int n_in,
                              void* d_out, int out_size, void* d_ws, size_t ws_size,
                              hipStream_t stream) {
    const float* x     = (const float*)d_in[0];   // (256,3,224,224) f32
    const int*   shift = (const int*)d_in[1];     // (256,2) i32
    float*       out   = (float*)d_out;           // (256,3,224,224) f32

    // 224 rows / (32 rows per block) = 7 blocks in x; channels in y; images in z.
    dim3 grid(HH / 32, C_CH, N_IMG);              // (7, 3, 256)
    RandomShiftsAug_kernel<<<grid, dim3(256), 0, stream>>>(x, shift, out);
}